// Router_64089501991417
// MI455X (gfx1250) — compile-verified
//
#include <hip/hip_runtime.h>
#include <hip/hip_bf16.h>
#include <math.h>

typedef float v2f __attribute__((ext_vector_type(2)));
typedef float v8f __attribute__((ext_vector_type(8)));

#define T_TOK   32768
#define D_IN    1024
#define H_DIM   512
#define E_EXP   64
#define ROWS    32         // tokens per block
#define CK      128        // K chunk for x staging
#define XS      132        // x LDS stride (mod 64 == 4 -> conflict-free)
#define HTS     33         // transposed-h stride (rows 32 + 1 pad)
#define CAPACITY 1280u     // 1.25 * T * K / E

// LDS floats: hT region (8 waves * 64 cols * 33) aliases x-chunk & partials
#define SM_HT    (8 * 64 * HTS)        // 16896 floats (also >= 32*132 x chunk, >= 8*2048 partials)
#define SM_LOG   (ROWS * 64)           // 2048
#define SM_STAT  64
#define SM_TOT   (SM_HT + SM_LOG + SM_STAT)   // 19008 floats = 76032 B

// ---------------------------------------------------------------------------
// Kernel 0: zero global accumulators (counts, usage, zsum)
// ---------------------------------------------------------------------------
__global__ void k_zero(unsigned int* counts, float* usage, float* zsum) {
    int t = threadIdx.x;
    if (t < 64) { counts[t] = 0u; usage[t] = 0.f; }
    if (t == 0) zsum[0] = 0.f;
}

// ---------------------------------------------------------------------------
// Kernel 1: fused router core. 32 tokens per block, 256 threads (8 waves).
// Wave wv owns h columns [wv*64, wv*64+64) for all 32 rows (2 M-tiles).
// ---------------------------------------------------------------------------
__global__ __launch_bounds__(256) void k_router(
    const float* __restrict__ x, const float* __restrict__ W1,
    const float* __restrict__ ln_g, const float* __restrict__ ln_b,
    const float* __restrict__ W2, float* __restrict__ out_rw,
    int* __restrict__ ti0, int* __restrict__ ti1,
    float* __restrict__ tw0, float* __restrict__ tw1,
    float* __restrict__ zsum)
{
    __shared__ float smem[SM_TOT];
    float* MAIN   = smem;                 // x chunk -> hT slices -> GEMM2 partials
    float* LOGITS = smem + SM_HT;         // 32 x 64
    float* STAT   = LOGITS + SM_LOG;      // 32 rows x {sum, sumsq}

    const int tid  = threadIdx.x;
    const int lane = tid & 31;
    const int wv   = tid >> 5;        // wave 0..7
    const int l16  = lane & 15;
    const int hi   = lane >> 4;       // 0 or 1
    const int row0 = blockIdx.x * ROWS;

    // ---- GEMM1 accumulators: 2 M-tiles x 4 N-tiles ----
    v8f acc[2][4];
    #pragma unroll
    for (int m = 0; m < 2; ++m)
        #pragma unroll
        for (int t = 0; t < 4; ++t)
            #pragma unroll
            for (int r = 0; r < 8; ++r) acc[m][t][r] = 0.f;

    for (int c = 0; c < D_IN / CK; ++c) {
        // stage x[row0..row0+32)[c*128 .. +128) into LDS, coalesced float4
        #pragma unroll
        for (int i = 0; i < 4; ++i) {
            int e  = tid + i * 256;          // 0..1023 float4 slots
            int r  = e >> 5;                 // row 0..31
            int c4 = e & 31;                 // float4 col
            float4 v = *(const float4*)(x + (size_t)(row0 + r) * D_IN + c * CK + c4 * 4);
            *(float4*)(MAIN + r * XS + c4 * 4) = v;
        }
        __syncthreads();

        for (int kk = 0; kk < CK; kk += 4) {
            // A frags: lanes 0-15 hold K=kk{+0,+1}, lanes 16-31 K=kk{+2,+3}
            v2f a0 = *(const v2f*)(MAIN + l16 * XS + kk + 2 * hi);
            v2f a1 = *(const v2f*)(MAIN + (16 + l16) * XS + kk + 2 * hi);
            int kg = c * CK + kk + 2 * hi;
            #pragma unroll
            for (int t = 0; t < 4; ++t) {
                int n = wv * 64 + t * 16 + l16;
                v2f b;
                b.x = W1[(size_t)kg * H_DIM + n];
                b.y = W1[(size_t)(kg + 1) * H_DIM + n];
                acc[0][t] = __builtin_amdgcn_wmma_f32_16x16x4_f32(
                    false, a0, false, b, (short)0, acc[0][t], false, false);
                acc[1][t] = __builtin_amdgcn_wmma_f32_16x16x4_f32(
                    false, a1, false, b, (short)0, acc[1][t], false, false);
            }
        }
        __syncthreads();
    }

    // ---- LayerNorm row statistics via shared atomics ----
    if (tid < 64) STAT[tid] = 0.f;
    __syncthreads();
    #pragma unroll
    for (int m = 0; m < 2; ++m)
        #pragma unroll
        for (int r = 0; r < 8; ++r) {
            float s = 0.f, q = 0.f;
            #pragma unroll
            for (int t = 0; t < 4; ++t) { float v = acc[m][t][r]; s += v; q += v * v; }
            int row = m * 16 + r + 8 * hi;
            atomicAdd(&STAT[2 * row],     s);
            atomicAdd(&STAT[2 * row + 1], q);
        }
    __syncthreads();

    // ---- normalize + affine + ReLU, write transposed h slice into MAIN ----
    float* HT = MAIN + wv * (64 * HTS);   // this wave's 64 cols x 32 rows (stride 33)
    #pragma unroll
    for (int m = 0; m < 2; ++m)
        #pragma unroll
        for (int r = 0; r < 8; ++r) {
            int row = m * 16 + r + 8 * hi;
            float mu  = STAT[2 * row] * (1.f / 512.f);
            float var = STAT[2 * row + 1] * (1.f / 512.f) - mu * mu;
            float rs  = rsqrtf(var + 1e-5f);
            #pragma unroll
            for (int t = 0; t < 4; ++t) {
                int col = wv * 64 + t * 16 + l16;
                float v = (acc[m][t][r] - mu) * rs * ln_g[col] + ln_b[col];
                HT[(t * 16 + l16) * HTS + row] = fmaxf(v, 0.f);
            }
        }
    __syncthreads();

    // ---- GEMM2: logits = h @ W2, split-K across 8 waves (64 K each) ----
    v8f acc2[2][4];
    #pragma unroll
    for (int m = 0; m < 2; ++m)
        #pragma unroll
        for (int t = 0; t < 4; ++t)
            #pragma unroll
            for (int r = 0; r < 8; ++r) acc2[m][t][r] = 0.f;

    const int ks = wv * 64;               // this wave's K slice = its own h cols
    for (int kk = 0; kk < 64; kk += 4) {
        int kl = kk + 2 * hi;             // local k within slice
        v2f a0, a1;
        a0.x = HT[(kl    ) * HTS + l16];
        a0.y = HT[(kl + 1) * HTS + l16];
        a1.x = HT[(kl    ) * HTS + 16 + l16];
        a1.y = HT[(kl + 1) * HTS + 16 + l16];
        int kg = ks + kl;
        #pragma unroll
        for (int t = 0; t < 4; ++t) {
            int n = t * 16 + l16;
            v2f b;
            b.x = W2[kg * E_EXP + n];
            b.y = W2[(kg + 1) * E_EXP + n];
            acc2[0][t] = __builtin_amdgcn_wmma_f32_16x16x4_f32(
                false, a0, false, b, (short)0, acc2[0][t], false, false);
            acc2[1][t] = __builtin_amdgcn_wmma_f32_16x16x4_f32(
                false, a1, false, b, (short)0, acc2[1][t], false, false);
        }
    }
    __syncthreads();   // all hT reads done -> safe to alias MAIN with partials

    #pragma unroll
    for (int m = 0; m < 2; ++m)
        #pragma unroll
        for (int t = 0; t < 4; ++t)
            #pragma unroll
            for (int r = 0; r < 8; ++r) {
                int row = m * 16 + r + 8 * hi, col = t * 16 + l16;
                MAIN[wv * 2048 + row * 64 + col] = acc2[m][t][r];
            }
    __syncthreads();

    for (int i = tid; i < ROWS * 64; i += 256) {
        float s = 0.f;
        #pragma unroll
        for (int w = 0; w < 8; ++w) s += MAIN[w * 2048 + i];
        LOGITS[i] = s;
    }
    __syncthreads();

    // ---- per-row epilogue: one thread per token row ----
    if (tid < ROWS) {
        float* Lr = LOGITS + tid * 64;
        // clip +-50 and L2 norm
        float ss = 0.f;
        for (int j = 0; j < 64; ++j) {
            float v = fminf(fmaxf(Lr[j], -50.f), 50.f);
            Lr[j] = v; ss += v * v;
        }
        float inv = 1.f / (fmaxf(sqrtf(ss), 1e-6f) + 1e-6f);
        float ml = -3.4e38f;
        for (int j = 0; j < 64; ++j) { float v = Lr[j] * inv; Lr[j] = v; ml = fmaxf(ml, v); }

        // softmax with temperature 0.1 (x10)
        float es = 0.f;
        for (int j = 0; j < 64; ++j) es += expf((Lr[j] - ml) * 10.f);
        float ies = 1.f / es;

        const int tg = row0 + tid;
        float best0 = -1.f, best1 = -1.f; int i0 = 0, i1 = 0;
        for (int j = 0; j < 64; ++j) {
            float p = expf((Lr[j] - ml) * 10.f) * ies;
            out_rw[(size_t)tg * 64 + j] = p;
            if (p > best0)      { best1 = best0; i1 = i0; best0 = p; i0 = j; }
            else if (p > best1) { best1 = p; i1 = j; }
        }
        float w0 = fminf(fmaxf(best0, 0.001f), 0.8f);
        float w1 = fminf(fmaxf(best1, 0.001f), 0.8f);
        float dd = fmaxf(w0 + w1, 1e-6f) + 1e-6f;
        ti0[tg] = i0; ti1[tg] = i1;
        tw0[tg] = w0 / dd; tw1[tg] = w1 / dd;

        // z-loss partial: log_softmax on (normalized) logits, no temperature
        float es2 = 0.f;
        for (int j = 0; j < 64; ++j) es2 += expf(Lr[j] - ml);
        float lse = ml + logf(es2);
        float zq = 0.f;
        for (int j = 0; j < 64; ++j) {
            float lp = fminf(fmaxf(Lr[j] - lse, -50.f), 50.f);
            zq += lp * lp;
        }
        atomicAdd(zsum, zq);
    }
}

// ---------------------------------------------------------------------------
// Kernel 2: histogram of top-1 expert (pass-0 counts)
// ---------------------------------------------------------------------------
__global__ __launch_bounds__(256) void k_hist(const int* __restrict__ ti0,
                                              unsigned int* __restrict__ counts) {
    __shared__ unsigned int hc[64];
    int tid = threadIdx.x;
    if (tid < 64) hc[tid] = 0u;
    __syncthreads();
    int t = blockIdx.x * 256 + tid;
    atomicAdd(&hc[ti0[t]], 1u);
    __syncthreads();
    if (tid < 64 && hc[tid]) atomicAdd(&counts[tid], hc[tid]);
}

// ---------------------------------------------------------------------------
// Kernel 3: dispatch mask (pass 1), ndm rows, usage accumulation
// ---------------------------------------------------------------------------
__global__ __launch_bounds__(256) void k_ndm(
    const int* __restrict__ ti0, const int* __restrict__ ti1,
    const float* __restrict__ tw0, const float* __restrict__ tw1,
    const unsigned int* __restrict__ counts,
    float* __restrict__ ndm, float* __restrict__ usage) {
    __shared__ float ul[64];
    int tid = threadIdx.x;
    if (tid < 64) ul[tid] = 0.f;
    __syncthreads();
    int t = blockIdx.x * 256 + tid;
    int e0 = ti0[t], e1 = ti1[t];
    float w0 = tw0[t], w1 = tw1[t];
    bool m = counts[e1] < CAPACITY;           // mask uses completed top-1 counts
    float den = fmaxf(w0 + (m ? w1 : 0.f), 1e-12f);
    float v0 = w0 / den;
    float v1 = m ? (w1 / den) : 0.f;
    float* rowp = ndm + (size_t)t * 64;
    float4 z4 = make_float4(0.f, 0.f, 0.f, 0.f);
    #pragma unroll
    for (int i = 0; i < 16; ++i) *(float4*)(rowp + i * 4) = z4;
    rowp[e0] = v0;
    if (m) rowp[e1] = v1;
    atomicAdd(&ul[e0], v0);
    if (m) atomicAdd(&ul[e1], v1);
    __syncthreads();
    if (tid < 64) atomicAdd(&usage[tid], ul[tid]);
}

// ---------------------------------------------------------------------------
// Kernel 4: finalize scalar loss
// ---------------------------------------------------------------------------
__global__ void k_loss(const float* __restrict__ usage,
                       const float* __restrict__ zsum,
                       float* __restrict__ out_loss) {
    __shared__ float red[64];
    int t = threadIdx.x;   // 64 threads
    float u = usage[t];
    red[t] = u;
    __syncthreads();
    for (int s = 32; s > 0; s >>= 1) { if (t < s) red[t] += red[t + s]; __syncthreads(); }
    float total = red[0];
    __syncthreads();
    float un = u / (fmaxf(total, 1e-6f) + 1e-6f);
    un = fminf(fmaxf(un, 1e-6f), 1.f);
    const float target = 1.f / 64.f;
    red[t] = target * (logf(target) - logf(un + 1e-6f));
    __syncthreads();
    for (int s = 32; s > 0; s >>= 1) { if (t < s) red[t] += red[t + s]; __syncthreads(); }
    if (t == 0) {
        float kl = fminf(red[0] / 64.f, 100.f);
        float z  = fminf(zsum[0] / (32768.f * 64.f), 100.f);
        float loss = 1e-4f * z + 1e-4f * kl;
        if (isnan(loss) || isinf(loss)) loss = 0.1f;
        out_loss[0] = loss;
    }
}

// ---------------------------------------------------------------------------
extern "C" void kernel_launch(void* const* d_in, const int* in_sizes, int n_in,
                              void* d_out, int out_size, void* d_ws, size_t ws_size,
                              hipStream_t stream) {
    (void)in_sizes; (void)n_in; (void)out_size; (void)ws_size;
    const float* x    = (const float*)d_in[0];
    const float* W1   = (const float*)d_in[1];
    const float* ln_g = (const float*)d_in[2];
    const float* ln_b = (const float*)d_in[3];
    const float* W2   = (const float*)d_in[4];

    float* out  = (float*)d_out;
    float* rw   = out;                           // [T,64]
    float* ndm  = out + (size_t)T_TOK * 64;      // [T,64]
    float* loss = out + (size_t)2 * T_TOK * 64;  // [1]

    int*          ti0    = (int*)d_ws;
    int*          ti1    = ti0 + T_TOK;
    float*        tw0    = (float*)(ti1 + T_TOK);
    float*        tw1    = tw0 + T_TOK;
    unsigned int* counts = (unsigned int*)(tw1 + T_TOK);
    float*        usage  = (float*)(counts + 64);
    float*        zsum   = usage + 64;

    k_zero  <<<1, 64, 0, stream>>>(counts, usage, zsum);
    k_router<<<T_TOK / ROWS, 256, 0, stream>>>(x, W1, ln_g, ln_b, W2,
                                               rw, ti0, ti1, tw0, tw1, zsum);
    k_hist  <<<T_TOK / 256, 256, 0, stream>>>(ti0, counts);
    k_ndm   <<<T_TOK / 256, 256, 0, stream>>>(ti0, ti1, tw0, tw1, counts, ndm, usage);
    k_loss  <<<1, 64, 0, stream>>>(usage, zsum, loss);
}